// Augment_Learner_34050500723299
// MI455X (gfx1250) — compile-verified
//
#include <hip/hip_runtime.h>

// Edge-scoring MLP: out[e] = relu(concat(emb[src[e]], emb[dst[e]]) @ W1 + b1) @ W2 + b2
//
// MI455X reasoning: nodes_emb (38.4 MB) is L2-resident (192 MB L2), so the
// ~1 GB of random per-edge gather traffic is served by L2, and HBM moves only
// ~62 MB. Compute is 32.8 GFLOP fp32. We keep full f32 precision (matches the
// reference bit-for-bit in rounding behavior) using V_WMMA_F32_16X16X4_F32:
// per 16-edge tile a wave issues 128 wmma ops (K=128 concat, N=64, 4 K per op).
// W1 is pre-swizzled into LDS in the exact B-fragment lane layout so each wmma
// consumes one ds_load_b64.

typedef __attribute__((ext_vector_type(2))) float v2f;
typedef __attribute__((ext_vector_type(8))) float v8f;

#define DIM 64
#define KK_TOT 32   // 128 K-values / 4 per wmma
#define NT 4        // 4 tiles of 16 output columns -> 64

__global__ __launch_bounds__(256) void edge_mlp_wmma(
    const float* __restrict__ nodes, const int* __restrict__ src,
    const int* __restrict__ dst, const float* __restrict__ W1,
    const float* __restrict__ b1, const float* __restrict__ W2,
    const float* __restrict__ b2, float* __restrict__ out, int E)
{
  // B fragments: frag f = n*32+kk; per lane l: (v0,v1) = W1[4kk+2*(l>>4) + {0,1}][n*16 + (l&15)]
  __shared__ v2f ldsB[KK_TOT * NT * 32];  // 128 frags * 32 lanes * 8B = 32 KB

  for (int g = threadIdx.x; g < KK_TOT * NT * 32; g += blockDim.x) {
    int f = g >> 5, ln = g & 31;
    int n = f >> 5, kk = f & 31;
    int hf = ln >> 4, c = (n << 4) + (ln & 15);
    int row = (kk << 2) + (hf << 1);
    v2f v;
    v.x = W1[row * DIM + c];
    v.y = W1[(row + 1) * DIM + c];
    ldsB[g] = v;
  }
  __syncthreads();

  const int lane = threadIdx.x & 31;
  const int half = lane >> 4;     // 0: rows M=0..7 / K lo pair, 1: rows M=8..15 / K hi pair
  const int l15  = lane & 15;     // output column within 16-wide tile

  float bias[NT], w2v[NT];
#pragma unroll
  for (int n = 0; n < NT; ++n) {
    bias[n] = b1[n * 16 + l15];
    w2v[n]  = W2[n * 16 + l15];
  }
  const float b2v = b2[0];

  const int wavesPerBlk = blockDim.x >> 5;
  const int gw = blockIdx.x * wavesPerBlk + (threadIdx.x >> 5);
  const int nw = gridDim.x * wavesPerBlk;
  const int tiles = (E + 15) >> 4;

  for (int t = gw; t < tiles; t += nw) {
    const int e0 = t << 4;
    int e = e0 + l15; if (e >= E) e = E - 1;   // clamp keeps EXEC all-ones through wmma
    const float* sp = nodes + (long)src[e] * DIM + (half << 1);
    const float* dp = nodes + (long)dst[e] * DIM + (half << 1);

    // A fragments, ISA f32 16x4 layout: lane<16 -> K=4kk..4kk+1, lane>=16 -> 4kk+2..4kk+3
    v2f aS[16], aD[16];
#pragma unroll
    for (int kk = 0; kk < 16; ++kk) {
      aS[kk] = *(const v2f*)(sp + (kk << 2));
      aD[kk] = *(const v2f*)(dp + (kk << 2));
    }

    float pl[8];
#pragma unroll
    for (int r = 0; r < 8; ++r) pl[r] = 0.f;

#pragma unroll
    for (int n = 0; n < NT; ++n) {
      v8f acc;
#pragma unroll
      for (int r = 0; r < 8; ++r) acc[r] = bias[n];   // C col = l15 -> bias per lane
#pragma unroll
      for (int kk = 0; kk < 16; ++kk) {               // src half of concat: K = 0..63
        v2f bfrag = ldsB[(n * 32 + kk) * 32 + lane];
        acc = __builtin_amdgcn_wmma_f32_16x16x4_f32(
            false, aS[kk], false, bfrag, (short)0, acc, false, false);
      }
#pragma unroll
      for (int kk = 0; kk < 16; ++kk) {               // dst half of concat: K = 64..127
        v2f bfrag = ldsB[(n * 32 + 16 + kk) * 32 + lane];
        acc = __builtin_amdgcn_wmma_f32_16x16x4_f32(
            false, aD[kk], false, bfrag, (short)0, acc, false, false);
      }
      // layer 2: relu then scale by W2[col], accumulate partial logit per row
#pragma unroll
      for (int r = 0; r < 8; ++r) {
        float h = acc[r] > 0.f ? acc[r] : 0.f;
        pl[r] += h * w2v[n];
      }
    }

    // sum over the 16 lanes (columns) of each half; xor offsets 1..8 stay in-half
#pragma unroll
    for (int r = 0; r < 8; ++r) {
#pragma unroll
      for (int off = 1; off < 16; off <<= 1)
        pl[r] += __shfl_xor(pl[r], off, 32);
      pl[r] += b2v;
    }

    // lane 0 holds logits for rows 0..7, lane 16 for rows 8..15
    if (e0 + 16 <= E) {
      if (l15 == 0) {
        float4* op = (float4*)(out + e0 + half * 8);
        op[0] = make_float4(pl[0], pl[1], pl[2], pl[3]);
        op[1] = make_float4(pl[4], pl[5], pl[6], pl[7]);
      }
    } else if (l15 == 0) {
#pragma unroll
      for (int r = 0; r < 8; ++r) {
        int idx = e0 + half * 8 + r;
        if (idx < E) out[idx] = pl[r];
      }
    }
  }
}

extern "C" void kernel_launch(void* const* d_in, const int* in_sizes, int n_in,
                              void* d_out, int out_size, void* d_ws, size_t ws_size,
                              hipStream_t stream) {
  const float* nodes = (const float*)d_in[0];
  const int*   src   = (const int*)d_in[1];
  const int*   dst   = (const int*)d_in[2];
  const float* W1    = (const float*)d_in[3];
  const float* b1    = (const float*)d_in[4];
  const float* W2    = (const float*)d_in[5];
  const float* b2    = (const float*)d_in[6];
  float* out = (float*)d_out;

  const int E = in_sizes[1];           // 2,000,000 edges
  const int tiles = (E + 15) / 16;     // 16 edges per wave-tile
  int blocks = (tiles + 7) / 8;        // 8 waves per 256-thread block
  if (blocks > 2048) blocks = 2048;    // grid-stride over tiles
  if (blocks < 1) blocks = 1;

  edge_mlp_wmma<<<blocks, 256, 0, stream>>>(nodes, src, dst, W1, b1, W2, b2, out, E);
}